// QuantumLayer_31559419691090
// MI455X (gfx1250) — compile-verified
//
#include <hip/hip_runtime.h>
#include <math.h>

// Quantum feature-map layer for MI455X (gfx1250, wave32).
//
//   out[b,q] = psi0(b)^T A_q psi0(b),   psi0 = tensor product of (cos,sin) of
//   half-angles from angles = x @ fm_w^T + fm_b;  A_q = Re(U^H Z_q U) fixed.
//
// Kernel 1 (qc_build_A): builds packed symmetric A_q (4 x 136 floats) in d_ws.
// Kernel 2 (qc_main):    WMMA f32 16x16x4 GEMM for angles + per-row quadratic
//                        forms; streams x once (memory-bound, ~68 MiB).

#define NQ   4
#define DIM  16
#define NF   64
#define RPB  256          // batch rows per block
#define NPAIR 136         // 16*17/2 packed symmetric entries

typedef float v2f __attribute__((ext_vector_type(2)));
typedef float v8f __attribute__((ext_vector_type(8)));

// ---------------------------------------------------------------------------
// Setup: evolve the 16 basis states through the fixed gate sequence to get U,
// then A_q[i][j] = sum_k z_q(k) * Re(conj(U[k][i]) U[k][j]).  Packed upper
// triangle, off-diagonals pre-multiplied by 2.
// ---------------------------------------------------------------------------
__global__ __launch_bounds__(256)
void qc_build_A(const float* __restrict__ qw, float* __restrict__ Apk)
{
    __shared__ float Ur[DIM][DIM];   // Ur[k][col]
    __shared__ float Ui[DIM][DIM];
    const int t = threadIdx.x;

    if (t < DIM) {
        float sr[DIM], si[DIM];
        #pragma unroll
        for (int k = 0; k < DIM; ++k) { sr[k] = (k == t) ? 1.f : 0.f; si[k] = 0.f; }

        auto rz = [&](float th, int q) {      // p = exp(-i th/2) on bit==0
            float s, c; __sincosf(0.5f * th, &s, &c);
            const int m = 1 << (3 - q);       // qubit 0 = MSB (reference layout)
            #pragma unroll
            for (int k = 0; k < DIM; ++k) {
                float pi = (k & m) ? s : -s;
                float nr = sr[k] * c - si[k] * pi;
                float ni = sr[k] * pi + si[k] * c;
                sr[k] = nr; si[k] = ni;
            }
        };
        auto ry = [&](float th, int q) {
            float s, c; __sincosf(0.5f * th, &s, &c);
            const int m = 1 << (3 - q);
            #pragma unroll
            for (int k = 0; k < DIM; ++k) {
                if (!(k & m)) {
                    const int k1 = k | m;
                    float r0 = sr[k], i0 = si[k], r1 = sr[k1], i1 = si[k1];
                    sr[k]  = c * r0 - s * r1;  si[k]  = c * i0 - s * i1;
                    sr[k1] = s * r0 + c * r1;  si[k1] = s * i0 + c * i1;
                }
            }
        };
        auto cnot = [&](int cq) {             // control cq, target cq+1
            const int cm = 1 << (3 - cq), tm = 1 << (2 - cq);
            #pragma unroll
            for (int k = 0; k < DIM; ++k) {
                if ((k & cm) && !(k & tm)) {
                    const int k1 = k | tm;
                    float tr = sr[k]; sr[k] = sr[k1]; sr[k1] = tr;
                    float ti = si[k]; si[k] = si[k1]; si[k1] = ti;
                }
            }
        };

        for (int q = 0; q < NQ; ++q)      { rz(qw[2*q],        q); ry(qw[2*q+1],        q); }
        for (int cq = 0; cq < NQ-1; ++cq)   cnot(cq);
        for (int q = 0; q < NQ; ++q)      { rz(qw[2*(q+NQ)],   q); ry(qw[2*(q+NQ)+1],   q); }

        #pragma unroll
        for (int k = 0; k < DIM; ++k) { Ur[k][t] = sr[k]; Ui[k][t] = si[k]; }
    }
    __syncthreads();

    const int i = t >> 4, j = t & 15;      // one (i,j) entry per thread
    float acc[NQ] = {0.f, 0.f, 0.f, 0.f};
    #pragma unroll
    for (int k = 0; k < DIM; ++k) {
        const float g = Ur[k][i] * Ur[k][j] + Ui[k][i] * Ui[k][j];
        #pragma unroll
        for (int q = 0; q < NQ; ++q)
            acc[q] += (((k >> (3 - q)) & 1) ? -g : g);
    }
    if (i <= j) {
        const int idx = i * 16 - (i * (i - 1)) / 2 + (j - i);   // packed index
        const float f = (i == j) ? 1.f : 2.f;
        #pragma unroll
        for (int q = 0; q < NQ; ++q) Apk[q * NPAIR + idx] = f * acc[q];
    }
}

// ---------------------------------------------------------------------------
// Main kernel: 256 threads = 8 waves; each wave does two 16-row WMMA tiles of
// the angles GEMM (f32 16x16x4, K=64 in 16 chunks), then one-thread-per-row
// circuit evaluation.
// ---------------------------------------------------------------------------
__global__ __launch_bounds__(256)
void qc_main(const float* __restrict__ x, const float* __restrict__ fm_w,
             const float* __restrict__ fm_b, const float* __restrict__ Apk,
             float* __restrict__ out)
{
    __shared__ float sAng[RPB][5];         // stride 5 -> conflict-free reads

    const int t    = threadIdx.x;
    const int lane = t & 31;
    const int wave = t >> 5;
    const int m    = lane & 15;            // A: row-in-tile; B: column (N)
    const int h    = lane >> 4;            // lane-half selector for K split

    const size_t rowBlock = (size_t)blockIdx.x * RPB;

    // B fragments: fm_w^T zero-padded to 16 columns, all 16 K-chunks.
    // Layout (ISA 7.12.2, 32-bit 16x4): v.x -> K=k0=4c+2h, v.y -> K=k0+1.
    v2f bfrag[16];
    #pragma unroll
    for (int c = 0; c < 16; ++c) {
        const int k0 = 4 * c + 2 * h;
        bfrag[c].x = (m < NQ) ? fm_w[m * NF + k0]     : 0.f;
        bfrag[c].y = (m < NQ) ? fm_w[m * NF + k0 + 1] : 0.f;
    }

    #pragma unroll
    for (int sTile = 0; sTile < 2; ++sTile) {
        const int tile = wave * 2 + sTile;                 // 16 rows per tile
        const float* xrow = x + (rowBlock + (size_t)tile * 16 + m) * NF;

        v8f acc = {};
        #pragma unroll
        for (int c = 0; c < 16; ++c) {
            v2f a;
            a.x = xrow[4 * c + 2 * h];
            a.y = xrow[4 * c + 2 * h + 1];
            acc = __builtin_amdgcn_wmma_f32_16x16x4_f32(
                      false, a, false, bfrag[c], (short)0, acc, false, false);
        }
        // C layout: lanes 0-15 hold M=r, lanes 16-31 hold M=r+8, N=lane&15.
        #pragma unroll
        for (int r = 0; r < 8; ++r) {
            const int rowInTile = h ? (r + 8) : r;
            if (m < NQ) sAng[tile * 16 + rowInTile][m] = acc[r];
        }
    }
    __syncthreads();

    // --------- per-thread: one batch row ---------
    float cq[NQ], sq[NQ];
    #pragma unroll
    for (int q = 0; q < NQ; ++q) {
        const float a = sAng[t][q] + fm_b[q];
        __sincosf(0.5f * a, &sq[q], &cq[q]);
    }
    const float e01[4] = { cq[0]*cq[1], cq[0]*sq[1], sq[0]*cq[1], sq[0]*sq[1] };
    const float e23[4] = { cq[2]*cq[3], cq[2]*sq[3], sq[2]*cq[3], sq[2]*sq[3] };
    float psi[DIM];
    #pragma unroll
    for (int i = 0; i < DIM; ++i) psi[i] = e01[i >> 2] * e23[i & 3];

    // Quadratic forms with packed symmetric A (uniform -> scalar loads):
    // out_q = sum_{i<=j} Apk[q][idx(i,j)] * psi_i * psi_j
    float o0 = 0.f, o1 = 0.f, o2 = 0.f, o3 = 0.f;
    #pragma unroll
    for (int i = 0; i < DIM; ++i) {
        #pragma unroll
        for (int j = i; j < DIM; ++j) {
            const int idx = i * 16 - (i * (i - 1)) / 2 + (j - i);
            const float pp = psi[i] * psi[j];
            o0 = fmaf(Apk[0 * NPAIR + idx], pp, o0);
            o1 = fmaf(Apk[1 * NPAIR + idx], pp, o1);
            o2 = fmaf(Apk[2 * NPAIR + idx], pp, o2);
            o3 = fmaf(Apk[3 * NPAIR + idx], pp, o3);
        }
    }
    ((float4*)out)[rowBlock + t] = make_float4(o0, o1, o2, o3);
}

// ---------------------------------------------------------------------------
extern "C" void kernel_launch(void* const* d_in, const int* in_sizes, int n_in,
                              void* d_out, int out_size, void* d_ws, size_t ws_size,
                              hipStream_t stream)
{
    const float* x    = (const float*)d_in[0];
    const float* fm_w = (const float*)d_in[1];
    const float* fm_b = (const float*)d_in[2];
    const float* qw   = (const float*)d_in[3];
    float*       out  = (float*)d_out;
    float*       Apk  = (float*)d_ws;        // 4*136 floats

    const int B = in_sizes[0] / NF;          // 262144

    qc_build_A<<<1, 256, 0, stream>>>(qw, Apk);
    qc_main<<<B / RPB, 256, 0, stream>>>(x, fm_w, fm_b, Apk, out);
}